// TFSamVisionAttention_43267500539978
// MI455X (gfx1250) — compile-verified
//
#include <hip/hip_runtime.h>
#include <hip/hip_bf16.h>

// ---------------------------------------------------------------------------
// SAM Vision Attention on MI455X (gfx1250), fp32 WMMA (exact wrt fp32 ref).
//   B=128, H=W=14 (HW=196), C=768, NH=12, HD=64, rel table L=27
// Pipeline:
//   1) QKV GEMM (25088x768 @ 768x2304) -> scatter to q/k/v [B*NH][196][64]
//   2) Fused attention per (b,head,qtile): S=scale*QK^T + relH + relW,
//      softmax, P@V  (all matmuls via v_wmma_f32_16x16x4_f32, LDS staged)
//   3) Proj GEMM (25088x768 @ 768x768) + bias -> d_out
// ---------------------------------------------------------------------------

typedef float v2f __attribute__((ext_vector_type(2)));
typedef float v8f __attribute__((ext_vector_type(8)));

#define B_      128
#define NH_     12
#define HD_     64
#define C_      768
#define HW_     196
#define BHN_    (B_ * NH_)          // 1536
#define HEADSZ  (HW_ * HD_)         // 12544
#define QKVSZ   (BHN_ * HEADSZ)     // 19267584 floats per q/k/v tensor

__device__ __forceinline__ v8f wmma_f32(v2f a, v2f b, v8f c) {
  // D = A(16x4) * B(4x16) + C(16x16), fp32, RNE — exact fp32 accumulate.
  return __builtin_amdgcn_wmma_f32_16x16x4_f32(false, a, false, b,
                                               (short)0, c, false, false);
}

// ---------------------------------------------------------------------------
// Generic fp32 WMMA GEMM: D = A(MxK) * B(KxN) + bias
// Block tile 128x128, K-tile 32. 256 threads = 8 waves, each wave 32(M)x64(N)
// = 2x4 grid of 16x16 accumulators.
// mode 0: row-major store D[m][n]
// mode 1: QKV scatter: col n -> (which,head,d), row m -> (b,p);
//         dst[which][(b*NH+head)*196 + p][d]
// ---------------------------------------------------------------------------
__global__ __launch_bounds__(256)
void gemm_f32_wmma(const float* __restrict__ A, const float* __restrict__ Bw,
                   const float* __restrict__ bias, float* __restrict__ Dst,
                   int M, int N, int K, int mode) {
  __shared__ v2f   As2[128][17];     // [m][k/2] pairs (A[m][2f], A[m][2f+1])
  __shared__ float Bs[32][128];      // [k][n]

  const int t    = threadIdx.x;
  const int wave = t >> 5;
  const int ln   = t & 31;
  const int l15  = ln & 15;
  const int hi   = ln >> 4;          // half-wave id
  const int wm   = wave & 3;         // 4 waves along M
  const int wn   = wave >> 2;        // 2 waves along N
  const int row0 = blockIdx.y * 128;
  const int col0 = blockIdx.x * 128;

  v8f acc[2][4] = {};

  for (int kt = 0; kt < K; kt += 32) {
    // A tile -> LDS (coalesced float2 reads along K)
#pragma unroll
    for (int i = 0; i < 8; i++) {
      int idx = i * 256 + t;
      int m = idx >> 4, f = idx & 15;
      As2[m][f] = *(const v2f*)(A + (size_t)(row0 + m) * K + kt + 2 * f);
    }
    // B tile -> LDS (coalesced float4 reads along N)
#pragma unroll
    for (int i = 0; i < 4; i++) {
      int idx = i * 256 + t;
      int k = idx >> 5, f4 = (idx & 31) * 4;
      *(float4*)&Bs[k][f4] =
          *(const float4*)(Bw + (size_t)(kt + k) * N + col0 + f4);
    }
    __syncthreads();

#pragma unroll
    for (int ks = 0; ks < 8; ks++) {
      const int k0 = ks * 4 + hi * 2;
      v2f a[2], b[4];
#pragma unroll
      for (int mt = 0; mt < 2; mt++)
        a[mt] = As2[wm * 32 + mt * 16 + l15][ks * 2 + hi];
#pragma unroll
      for (int nt = 0; nt < 4; nt++) {
        int n = wn * 64 + nt * 16 + l15;
        v2f bb; bb[0] = Bs[k0][n]; bb[1] = Bs[k0 + 1][n];
        b[nt] = bb;
      }
#pragma unroll
      for (int mt = 0; mt < 2; mt++)
#pragma unroll
        for (int nt = 0; nt < 4; nt++)
          acc[mt][nt] = wmma_f32(a[mt], b[nt], acc[mt][nt]);
    }
    __syncthreads();
  }

  // Epilogue (dims are exact multiples of the tile; no masking needed)
#pragma unroll
  for (int mt = 0; mt < 2; mt++) {
#pragma unroll
    for (int nt = 0; nt < 4; nt++) {
      const int n = col0 + wn * 64 + nt * 16 + l15;
      const float bv = bias[n];
#pragma unroll
      for (int r = 0; r < 8; r++) {
        const int m = row0 + wm * 32 + mt * 16 + r + 8 * hi;
        const float v = acc[mt][nt][r] + bv;
        if (mode == 0) {
          Dst[(size_t)m * N + n] = v;
        } else {
          const int which = n / C_;
          const int c = n - which * C_;
          const int head = c >> 6, d = c & 63;
          const int b = m / HW_, p = m - b * HW_;
          Dst[(size_t)which * QKVSZ +
              ((size_t)(b * NH_ + head) * HW_ + p) * HD_ + d] = v;
        }
      }
    }
  }
}

// ---------------------------------------------------------------------------
// Fused attention. Grid: (7 q-tiles of 32 rows, B*NH). Block: 256 threads.
// Keys padded 196 -> 224 (14 WMMA N-tiles); pad columns masked to -3e38.
// Dynamic LDS layout (floats):
//   Ks[224][68] Vs[224][64] Qs[32][68] Ss[32][225] Rhs[27][64] Rws[27][64]
//   relh[32][16] relw[32][16] red[32][8] rmax[32] rsum[32]   => 174976 B
// ---------------------------------------------------------------------------
#define ATT_SMEM_FLOATS (224*68 + 224*64 + 32*68 + 32*225 + 27*64*2 + \
                         32*16*2 + 32*8 + 32 + 32)

__global__ __launch_bounds__(256)
void sam_attention(const float* __restrict__ qkv,
                   const float* __restrict__ relph,
                   const float* __restrict__ relpw,
                   float* __restrict__ outb) {
  const int qt   = blockIdx.x;          // query tile: rows qt*32 .. qt*32+31
  const int bh   = blockIdx.y;          // 0..1535
  const int b    = bh / NH_;
  const int head = bh - b * NH_;
  const int t    = threadIdx.x;
  const int wave = t >> 5;
  const int ln   = t & 31;
  const int l15  = ln & 15;
  const int hi   = ln >> 4;

  extern __shared__ float smem[];
  float* Ks   = smem;                   // [224][68]
  float* Vs   = Ks + 224 * 68;          // [224][64]
  float* Qs   = Vs + 224 * 64;          // [32][68]
  float* Ss   = Qs + 32 * 68;           // [32][225]
  float* Rhs  = Ss + 32 * 225;          // [27][64]
  float* Rws  = Rhs + 27 * 64;          // [27][64]
  float* relh = Rws + 27 * 64;          // [32][16]
  float* relw = relh + 32 * 16;         // [32][16]
  float* red  = relw + 32 * 16;         // [32][8]
  float* rmax = red + 32 * 8;           // [32]
  float* rsum = rmax + 32;              // [32]

  const float* qh_ = qkv + 0 * (size_t)QKVSZ + (size_t)bh * HEADSZ;
  const float* kh_ = qkv + 1 * (size_t)QKVSZ + (size_t)bh * HEADSZ;
  const float* vh_ = qkv + 2 * (size_t)QKVSZ + (size_t)bh * HEADSZ;

  // ---- stage K, V (zero-pad keys >= 196) ----
#pragma unroll
  for (int i = 0; i < 14; i++) {
    int idx = i * 256 + t;
    int key = idx >> 4, f4 = (idx & 15) * 4;
    float4 kv = make_float4(0.f, 0.f, 0.f, 0.f);
    float4 vv = make_float4(0.f, 0.f, 0.f, 0.f);
    if (key < HW_) {
      kv = *(const float4*)(kh_ + key * HD_ + f4);
      vv = *(const float4*)(vh_ + key * HD_ + f4);
    }
    *(float4*)(Ks + key * 68 + f4) = kv;
    *(float4*)(Vs + key * 64 + f4) = vv;
  }
  // ---- stage Q tile (zero-pad queries >= 196) ----
#pragma unroll
  for (int i = 0; i < 2; i++) {
    int idx = i * 256 + t;
    int m = idx >> 4, f4 = (idx & 15) * 4;
    int p = qt * 32 + m;
    float4 qv = make_float4(0.f, 0.f, 0.f, 0.f);
    if (p < HW_) qv = *(const float4*)(qh_ + p * HD_ + f4);
    *(float4*)(Qs + m * 68 + f4) = qv;
  }
  // ---- stage rel-pos tables ----
  for (int idx = t; idx < 27 * 64; idx += 256) {
    Rhs[idx] = relph[idx];
    Rws[idx] = relpw[idx];
  }
  __syncthreads();

  // ---- rel_h / rel_w per query: 32 rows x (14 + 14) dot-64 products ----
  for (int task = t; task < 32 * 28; task += 256) {
    int m = task / 28, j = task - m * 28;
    int p = qt * 32 + m;
    const float* qrow = Qs + m * 68;
    float s = 0.f;
    if (j < 14) {
      int qh = p / 14;
      int ir = qh - j + 13; ir = ir < 0 ? 0 : (ir > 26 ? 26 : ir);
      const float* rr = Rhs + ir * 64;
      for (int c = 0; c < 64; c++) s += qrow[c] * rr[c];
      relh[m * 16 + j] = s;
    } else {
      int jw = j - 14;
      int qw = p % 14;
      int ir = qw - jw + 13; ir = ir < 0 ? 0 : (ir > 26 ? 26 : ir);
      const float* rr = Rws + ir * 64;
      for (int c = 0; c < 64; c++) s += qrow[c] * rr[c];
      relw[m * 16 + jw] = s;
    }
  }
  __syncthreads();

  // ---- S = scale * Q @ K^T + rel, masked; waves split the 14 key tiles ----
  const float scale = 0.125f;  // 64^-0.5
  for (int nt = wave; nt < 14; nt += 8) {
    v8f accs[2] = {};
#pragma unroll
    for (int ks = 0; ks < 16; ks++) {
      const int k0 = ks * 4 + hi * 2;
      const int key = nt * 16 + l15;
      v2f bfr = *(const v2f*)(Ks + key * 68 + k0);   // B[k][n] = K[n][k]
#pragma unroll
      for (int mt = 0; mt < 2; mt++) {
        v2f afr = *(const v2f*)(Qs + (mt * 16 + l15) * 68 + k0);
        accs[mt] = wmma_f32(afr, bfr, accs[mt]);
      }
    }
    const int kk = nt * 16 + l15;
    const int khh = kk / 14, kww = kk - (kk / 14) * 14;
    const bool valid = kk < HW_;
#pragma unroll
    for (int mt = 0; mt < 2; mt++) {
#pragma unroll
      for (int r = 0; r < 8; r++) {
        const int m = mt * 16 + r + 8 * hi;
        float v = valid ? (scale * accs[mt][r] + relh[m * 16 + khh] +
                           relw[m * 16 + kww])
                        : -3.0e38f;
        Ss[m * 225 + kk] = v;
      }
    }
  }
  __syncthreads();

  // ---- softmax over 224 cols (pad cols -> exp == 0): 8 threads per row ----
  {
    const int row = t >> 3, sub = t & 7;
    float* srow = Ss + row * 225;
    float mx = -3.4e38f;
    for (int j = 0; j < 28; j++) mx = fmaxf(mx, srow[sub * 28 + j]);
    red[row * 8 + sub] = mx;
    __syncthreads();
    if (sub == 0) {
      float m2 = red[row * 8];
      for (int j = 1; j < 8; j++) m2 = fmaxf(m2, red[row * 8 + j]);
      rmax[row] = m2;
    }
    __syncthreads();
    const float m2 = rmax[row];
    float sum = 0.f;
    for (int j = 0; j < 28; j++) {
      float e = __expf(srow[sub * 28 + j] - m2);
      srow[sub * 28 + j] = e;
      sum += e;
    }
    red[row * 8 + sub] = sum;
    __syncthreads();
    if (sub == 0) {
      float s2 = 0.f;
      for (int j = 0; j < 8; j++) s2 += red[row * 8 + j];
      rsum[row] = s2;
    }
    __syncthreads();
    const float inv = 1.0f / rsum[row];
    for (int j = 0; j < 28; j++) srow[sub * 28 + j] *= inv;
  }
  __syncthreads();

  // ---- O = P @ V : each wave owns one 16x16 output tile (2M x 4N) ----
  {
    const int mt = wave >> 2, nt = wave & 3;
    v8f acc = {};
    for (int ks = 0; ks < 56; ks++) {
      const int k0 = ks * 4 + hi * 2;
      v2f afr;
      afr[0] = Ss[(mt * 16 + l15) * 225 + k0];
      afr[1] = Ss[(mt * 16 + l15) * 225 + k0 + 1];
      v2f bfr;
      const int d = nt * 16 + l15;
      bfr[0] = Vs[k0 * 64 + d];
      bfr[1] = Vs[(k0 + 1) * 64 + d];
      acc = wmma_f32(afr, bfr, acc);
    }
    const int d = nt * 16 + l15;
#pragma unroll
    for (int r = 0; r < 8; r++) {
      const int m = mt * 16 + r + 8 * hi;
      const int p = qt * 32 + m;
      if (p < HW_)
        outb[((size_t)(b * HW_ + p)) * C_ + head * HD_ + d] = acc[r];
    }
  }
}

// ---------------------------------------------------------------------------
extern "C" void kernel_launch(void* const* d_in, const int* in_sizes, int n_in,
                              void* d_out, int out_size, void* d_ws,
                              size_t ws_size, hipStream_t stream) {
  const float* x    = (const float*)d_in[0];  // (128,14,14,768) = (25088,768)
  const float* wqkv = (const float*)d_in[1];  // (768,2304)
  const float* bqkv = (const float*)d_in[2];  // (2304,)
  const float* rph  = (const float*)d_in[3];  // (27,64)
  const float* rpw  = (const float*)d_in[4];  // (27,64)
  const float* wp   = (const float*)d_in[5];  // (768,768)
  const float* bp   = (const float*)d_in[6];  // (768,)

  float* ws       = (float*)d_ws;
  float* qkv      = ws;                         // 3 * 19267584 floats
  float* attn_out = ws + 3 * (size_t)QKVSZ;     // 19267584 floats
  float* out      = (float*)d_out;

  // 1) QKV projection: M=25088, N=2304, K=768 -> scatter to q/k/v heads
  dim3 g1(2304 / 128, 25088 / 128);
  gemm_f32_wmma<<<g1, 256, 0, stream>>>(x, wqkv, bqkv, qkv, 25088, 2304, 768, 1);

  // 2) fused attention (174976 B dynamic LDS; WGP budget is 320 KB)
  const size_t smem = (size_t)ATT_SMEM_FLOATS * sizeof(float);
  hipFuncSetAttribute((const void*)sam_attention,
                      hipFuncAttributeMaxDynamicSharedMemorySize, (int)smem);
  dim3 g2(7, BHN_);
  sam_attention<<<g2, 256, smem, stream>>>(qkv, rph, rpw, attn_out);

  // 3) output projection: M=25088, N=768, K=768 -> d_out
  dim3 g3(768 / 128, 25088 / 128);
  gemm_f32_wmma<<<g3, 256, 0, stream>>>(attn_out, wp, bp, out, 25088, 768, 768, 0);
}